// MyRNN_37495064494172
// MI455X (gfx1250) — compile-verified
//
#include <hip/hip_runtime.h>
#include <math.h>

typedef __attribute__((ext_vector_type(2))) float v2f;
typedef __attribute__((ext_vector_type(8))) float v8f;
typedef __attribute__((ext_vector_type(4))) float f4v;

#define B_  1024
#define T_  80
#define E_  128
#define U_  128
#define ROWPAD 132   // LDS row stride in floats: keeps b64 A-fragment reads bank-conflict-free

__global__ __launch_bounds__(256)
void rnn_fused_wmma_f32(const int*   __restrict__ tokens,
                        const float* __restrict__ emb,
                        const float* __restrict__ Wx,
                        const float* __restrict__ Wh,
                        const float* __restrict__ bias,
                        const float* __restrict__ fcw,
                        const float* __restrict__ fcb,
                        float*       __restrict__ out)
{
    __shared__ float hb[16 * ROWPAD];        // h state   [16 rows][128 cols]
    __shared__ float xb[2][16 * ROWPAD];     // x tile, double buffered

    const int tid  = threadIdx.x;
    const int wave = tid >> 5;               // 0..7  -> N-tile of 16 columns
    const int lane = tid & 31;
    const int n    = lane & 15;              // column within tile / A-row index
    const int kb   = (lane >> 4) << 1;       // K sub-offset 0 or 2 (per 16x4 f32 layout)
    const int bb   = blockIdx.x;             // which 16-row batch block
    const int col  = wave * 16 + n;          // global output column 0..127

    // ---- preload B fragments for Wx and Wh into VGPRs (constant across all 80 steps) ----
    // mirror of documented 32-bit A 16x4 layout: VGPR0 holds K=kb, VGPR1 holds K=kb+1
    v2f bx[32], bw[32];
#pragma unroll
    for (int ks = 0; ks < 32; ++ks) {
        const int k = 4 * ks + kb;
        bx[ks].x = Wx[(k    ) * U_ + col];
        bx[ks].y = Wx[(k + 1) * U_ + col];
        bw[ks].x = Wh[(k    ) * U_ + col];
        bw[ks].y = Wh[(k + 1) * U_ + col];
    }
    const float bias_c = bias[col];

    // ---- zero h0 ----
    for (int i = tid; i < 16 * ROWPAD; i += 256) hb[i] = 0.0f;

    // ---- stage x tile for t = 0 (gathered embedding rows) ----
    for (int idx = tid; idx < 512; idx += 256) {
        const int r   = idx >> 5;            // row 0..15
        const int c4  = idx & 31;            // float4 chunk 0..31
        const int tok = tokens[(bb * 16 + r) * T_ + 0];
        const f4v v   = *(const f4v*)(emb + (long)tok * E_ + c4 * 4);
        const int e   = r * ROWPAD + c4 * 4;
        xb[0][e + 0] = v.x; xb[0][e + 1] = v.y;
        xb[0][e + 2] = v.z; xb[0][e + 3] = v.w;
    }
    __syncthreads();

    const int m = n;                         // A-fragment row for this lane

    for (int t = 0; t < T_; ++t) {
        // ---- issue gather of NEXT timestep's embedding rows into registers early ----
        f4v pf0 = {}, pf1 = {};
        if (t + 1 < T_) {
            {
                const int r = tid >> 5, c4 = tid & 31;
                const int tok = tokens[(bb * 16 + r) * T_ + (t + 1)];
                pf0 = *(const f4v*)(emb + (long)tok * E_ + c4 * 4);
            }
            {
                const int idx = tid + 256;
                const int r = idx >> 5, c4 = idx & 31;
                const int tok = tokens[(bb * 16 + r) * T_ + (t + 1)];
                pf1 = *(const f4v*)(emb + (long)tok * E_ + c4 * 4);
            }
        }

        // ---- C = bias; C += x_t @ Wx ; C += h @ Wh  (full f32 WMMA) ----
        const float* xcur = xb[t & 1];
        v8f c;
#pragma unroll
        for (int i = 0; i < 8; ++i) c[i] = bias_c;

#pragma unroll
        for (int ks = 0; ks < 32; ++ks) {
            v2f a = *(const v2f*)&xcur[m * ROWPAD + 4 * ks + kb];
            c = __builtin_amdgcn_wmma_f32_16x16x4_f32(
                    false, a, false, bx[ks], (short)0, c, false, false);
        }
#pragma unroll
        for (int ks = 0; ks < 32; ++ks) {
            v2f a = *(const v2f*)&hb[m * ROWPAD + 4 * ks + kb];
            c = __builtin_amdgcn_wmma_f32_16x16x4_f32(
                    false, a, false, bw[ks], (short)0, c, false, false);
        }

        __syncthreads();   // everyone done READING hb/xcur for step t

        // ---- h' = tanh(C): write back per C/D layout (VGPR i -> row i or i+8) ----
#pragma unroll
        for (int i = 0; i < 8; ++i) {
            const int row = i + ((lane >> 4) << 3);
            hb[row * ROWPAD + col] = tanhf(c[i]);
        }
        // ---- commit next x tile ----
        if (t + 1 < T_) {
            float* xnxt = xb[(t + 1) & 1];
            {
                const int r = tid >> 5, c4 = tid & 31;
                const int e = r * ROWPAD + c4 * 4;
                xnxt[e + 0] = pf0.x; xnxt[e + 1] = pf0.y;
                xnxt[e + 2] = pf0.z; xnxt[e + 3] = pf0.w;
            }
            {
                const int idx = tid + 256;
                const int r = idx >> 5, c4 = idx & 31;
                const int e = r * ROWPAD + c4 * 4;
                xnxt[e + 0] = pf1.x; xnxt[e + 1] = pf1.y;
                xnxt[e + 2] = pf1.z; xnxt[e + 3] = pf1.w;
            }
        }
        __syncthreads();   // hb / next x visible before step t+1 reads
    }

    // ---- logits = h_T @ fc_w + fc_b ; sigmoid ----  (16 rows per block; trivial cost)
    if (tid < 16) {
        float dot = fcb[0];
        for (int k = 0; k < U_; ++k) dot += hb[tid * ROWPAD + k] * fcw[k];
        out[bb * 16 + tid] = 1.0f / (1.0f + expf(-dot));
    }
}

extern "C" void kernel_launch(void* const* d_in, const int* in_sizes, int n_in,
                              void* d_out, int out_size, void* d_ws, size_t ws_size,
                              hipStream_t stream) {
    const int*   tokens = (const int*)  d_in[0];
    const float* emb    = (const float*)d_in[1];
    const float* Wx     = (const float*)d_in[2];
    const float* Wh     = (const float*)d_in[3];
    const float* b      = (const float*)d_in[4];
    const float* fcw    = (const float*)d_in[5];
    const float* fcb    = (const float*)d_in[6];
    float* out = (float*)d_out;

    dim3 grid(B_ / 16);   // 64 independent 16-row scan blocks
    dim3 block(256);      // 8 waves: one 16-col N-tile each
    rnn_fused_wmma_f32<<<grid, block, 0, stream>>>(tokens, emb, Wx, Wh, b, fcw, fcb, out);
}